// GNNCheckLayer_71614284693531
// MI455X (gfx1250) — compile-verified
//
#include <hip/hip_runtime.h>

#define BATCH 128
#define NN    32768
#define KNB   16

// ---------------------------------------------------------------------------
// Kernel 1: pack check_index + edge_type into one int per (node, neighbor).
//   valid  : packed = idx | (edge_type << 20)      (idx < 2^15, et in 0..3)
//   invalid: packed = 0 | (4 << 20)                (idx 0 is a safe gather;
//                                                   LUT entry 4 = {w=0,b=0}
//                                                   -> v = 0 exactly, which
//                                                   reproduces the reference
//                                                   invalid-edge semantics)
// ---------------------------------------------------------------------------
__global__ void gnn_pack_kernel(const int* __restrict__ ci,
                                const int* __restrict__ et,
                                int* __restrict__ packed) {
    int i = blockIdx.x * blockDim.x + threadIdx.x;
    if (i < NN * KNB) {
        int c = ci[i];
        packed[i] = (c < 0) ? (4 << 20) : (c | (et[i] << 20));
    }
}

// ---------------------------------------------------------------------------
// Kernel 2: transpose x (BATCH x NN) -> xt (NN x BATCH).
// Tile: 64 nodes x 128 batch in LDS (34 KB). Global loads AND stores are
// b128: a full xt row (128 floats = 512B) is exactly one wave of float4.
// ---------------------------------------------------------------------------
__global__ __launch_bounds__(256)
void gnn_transpose_kernel(const float* __restrict__ x, float* __restrict__ xt) {
    __shared__ float tile[64][133];          // [node][batch], padded
    const int tid = threadIdx.x;
    const int n0  = blockIdx.x * 64;

    // Load: float4 along the node dim, coalesced.
    {
        const int q  = tid & 15;             // 0..15 -> 64 floats of node dim
        const int b0 = tid >> 4;             // 0..15
        const float4* __restrict__ x4 = (const float4*)x;
#pragma unroll
        for (int p = 0; p < 8; ++p) {
            const int b = b0 + (p << 4);     // 0..127
            const float4 v = x4[((size_t)b * NN + n0) / 4 + q];
            tile[4 * q + 0][b] = v.x;
            tile[4 * q + 1][b] = v.y;
            tile[4 * q + 2][b] = v.z;
            tile[4 * q + 3][b] = v.w;
        }
    }
    __syncthreads();

    // Store: float4 along the batch dim (one wave = one whole xt row).
    {
        const int tx = tid & 31;             // float4 slot within 128-batch row
        const int ty = tid >> 5;             // 0..7
        float4* __restrict__ xt4 = (float4*)xt;
#pragma unroll
        for (int p = 0; p < 8; ++p) {
            const int nr = ty + (p << 3);    // 0..63
            float4 w;
            w.x = tile[nr][4 * tx + 0];
            w.y = tile[nr][4 * tx + 1];
            w.z = tile[nr][4 * tx + 2];
            w.w = tile[nr][4 * tx + 3];
            xt4[(size_t)(n0 + nr) * (BATCH / 4) + tx] = w;
        }
    }
}

// ---------------------------------------------------------------------------
// Kernel 3: main gather/reduce.
// One wave32 per node: each lane holds a float4 of batch values, so ONE
// global_load_b128 per (node, neighbor) covers the entire batch dim (512B,
// fully coalesced, L2-resident). Indices broadcast via __shfl. Branch-free
// inner loop: invalid edges map to LUT entry 4 ({0,0}) giving v == 0 exactly.
// Results staged in LDS (32 nodes x 128 batch), written as b128 stores.
// ---------------------------------------------------------------------------
__global__ __launch_bounds__(256)
void gnn_main_kernel(const float* __restrict__ xt,
                     const int* __restrict__ packed,
                     const float* __restrict__ W,
                     const float* __restrict__ Bv,
                     const float* __restrict__ alpha,
                     const float* __restrict__ beta,
                     float* __restrict__ out) {
    __shared__ float sW[8], sB[8];
    __shared__ __align__(16) float stile[32][132];   // [node][batch], 528B rows

    const int tid  = threadIdx.x;
    const int wid  = tid >> 5;
    const int lane = tid & 31;
    if (tid < 8) {
        sW[tid] = (tid < 4) ? W[tid]  : 0.0f;
        sB[tid] = (tid < 4) ? Bv[tid] : 0.0f;
    }
    __syncthreads();

    const float al = alpha[0];
    const float be = beta[0];
    const int n0 = blockIdx.x * 32;
    const float4* __restrict__ xt4 = (const float4*)xt;  // 32 float4 per node row

#pragma unroll
    for (int i = 0; i < 4; ++i) {
        const int n   = n0 + (wid << 2) + i;
        const int row = (wid << 2) + i;
        const int pk  = packed[n * KNB + (lane & 15)];   // one 64B line, mirrored

        unsigned ng0 = 0u, ng1 = 0u, ng2 = 0u, ng3 = 0u; // sign parity (bit 31)
        const float inf = __int_as_float(0x7f800000);
        float m0 = inf, m1 = inf, m2 = inf, m3 = inf;

#pragma unroll
        for (int k = 0; k < KNB; ++k) {
            const int   p   = __shfl(pk, k, 32);         // wave-uniform
            const int   idx = p & 0xFFFF;
            const int   e   = p >> 20;                   // 0..4
            const float w   = sW[e];
            const float bb  = sB[e];
            const float4 xv = xt4[(size_t)idx * 32 + lane];
            const float v0 = fmaf(xv.x, w, bb);
            const float v1 = fmaf(xv.y, w, bb);
            const float v2 = fmaf(xv.z, w, bb);
            const float v3 = fmaf(xv.w, w, bb);
            ng0 ^= __float_as_uint(v0 + 1e-10f);
            ng1 ^= __float_as_uint(v1 + 1e-10f);
            ng2 ^= __float_as_uint(v2 + 1e-10f);
            ng3 ^= __float_as_uint(v3 + 1e-10f);
            m0 = fminf(m0, (v0 == 0.0f) ? 1e10f : fabsf(v0));
            m1 = fminf(m1, (v1 == 0.0f) ? 1e10f : fabsf(v1));
            m2 = fminf(m2, (v2 == 0.0f) ? 1e10f : fabsf(v2));
            m3 = fminf(m3, (v3 == 0.0f) ? 1e10f : fabsf(v3));
        }

        float r0 = fmaf(al, m0, be);
        float r1 = fmaf(al, m1, be);
        float r2 = fmaf(al, m2, be);
        float r3 = fmaf(al, m3, be);
        // multiply by sign_product (+/-1) == conditional sign-bit flip
        r0 = __uint_as_float(__float_as_uint(r0) ^ (ng0 & 0x80000000u));
        r1 = __uint_as_float(__float_as_uint(r1) ^ (ng1 & 0x80000000u));
        r2 = __uint_as_float(__float_as_uint(r2) ^ (ng2 & 0x80000000u));
        r3 = __uint_as_float(__float_as_uint(r3) ^ (ng3 & 0x80000000u));

        float4 rr; rr.x = r0; rr.y = r1; rr.z = r2; rr.w = r3;
        *(float4*)&stile[row][lane * 4] = rr;     // 16B-aligned (row = 528B)
    }
    __syncthreads();

    // Store: b128 along nodes, coalesced 128B per wave per row.
#pragma unroll
    for (int p = 0; p < 4; ++p) {
        const int t  = tid + (p << 8);   // 0..1023
        const int q  = t & 7;            // node quad (0..7 -> 32 nodes)
        const int bb = t >> 3;           // batch row 0..127
        float4 r;
        r.x = stile[q * 4 + 0][bb];
        r.y = stile[q * 4 + 1][bb];
        r.z = stile[q * 4 + 2][bb];
        r.w = stile[q * 4 + 3][bb];
        *((float4*)(out + (size_t)bb * NN + n0) + q) = r;
    }
}

// ---------------------------------------------------------------------------
// Fallback (only if workspace is too small): direct gather, one thread per
// (batch, node). Coalesced output; gathers hit L2 (whole input is resident).
// ---------------------------------------------------------------------------
__global__ __launch_bounds__(256)
void gnn_fallback_kernel(const float* __restrict__ x,
                         const int* __restrict__ ci,
                         const int* __restrict__ et,
                         const float* __restrict__ W,
                         const float* __restrict__ Bv,
                         const float* __restrict__ alpha,
                         const float* __restrict__ beta,
                         float* __restrict__ out) {
    const int t = blockIdx.x * blockDim.x + threadIdx.x;
    if (t >= BATCH * NN) return;
    const int n = t % NN;
    const int b = t / NN;
    const float* xb = x + (size_t)b * NN;

    unsigned neg = 0u;
    float m = __int_as_float(0x7f800000);
#pragma unroll
    for (int k = 0; k < KNB; ++k) {
        const int c = ci[n * KNB + k];
        float v = 0.0f;
        if (c >= 0) {
            const int e = et[n * KNB + k];
            v = fmaf(xb[c], W[e], Bv[e]);
        }
        neg ^= __float_as_uint(v + 1e-10f);
        float a = fabsf(v);
        a = (a == 0.0f) ? 1e10f : a;
        m = fminf(m, a);
    }
    float r = fmaf(alpha[0], m, beta[0]);
    out[t] = __uint_as_float(__float_as_uint(r) ^ (neg & 0x80000000u));
}

extern "C" void kernel_launch(void* const* d_in, const int* in_sizes, int n_in,
                              void* d_out, int out_size, void* d_ws, size_t ws_size,
                              hipStream_t stream) {
    const float* x  = (const float*)d_in[0];
    const int*   ci = (const int*)d_in[1];   // check_index_tensor (32-bit per harness)
    const int*   et = (const int*)d_in[2];   // edge_type_tensor
    const float* W  = (const float*)d_in[3];
    const float* Bv = (const float*)d_in[4];
    const float* al = (const float*)d_in[5];
    const float* be = (const float*)d_in[6];
    float* out = (float*)d_out;

    const size_t xt_bytes   = (size_t)NN * BATCH * sizeof(float);   // 16 MB
    const size_t pack_bytes = (size_t)NN * KNB * sizeof(int);       //  2 MB

    if (ws_size >= xt_bytes + pack_bytes) {
        float* xt     = (float*)d_ws;
        int*   packed = (int*)((char*)d_ws + xt_bytes);

        gnn_pack_kernel<<<dim3((NN * KNB + 255) / 256), dim3(256), 0, stream>>>(ci, et, packed);
        gnn_transpose_kernel<<<dim3(NN / 64), dim3(256), 0, stream>>>(x, xt);
        gnn_main_kernel<<<dim3(NN / 32), dim3(256), 0, stream>>>(
            xt, packed, W, Bv, al, be, out);
    } else {
        gnn_fallback_kernel<<<dim3((BATCH * NN + 255) / 256), dim3(256), 0, stream>>>(
            x, ci, et, W, Bv, al, be, out);
    }
}